// MaxPooling_34677565948741
// MI455X (gfx1250) — compile-verified
//
#include <hip/hip_runtime.h>
#include <stdint.h>

// ---------------- problem constants (from reference) ----------------
#define GRIDV   64
#define NBATCH  16
#define SSEG    (NBATCH * GRIDV * GRIDV)   // 65536 clusters
#define FDIM    64
#define SORTCAP 2048                        // per-row LDS sort capacity
#define LDSROWS 128                         // feature rows staged per batch in cluster-max

// ---------------- async global->LDS helpers (gfx1250 ASYNCcnt path) --------
__device__ __forceinline__ void async_load_b128(uint32_t lds_addr, uint64_t gaddr) {
  asm volatile("global_load_async_to_lds_b128 %0, %1, off"
               :: "v"(lds_addr), "v"(gaddr) : "memory");
}
__device__ __forceinline__ void wait_async0() {
#if __has_builtin(__builtin_amdgcn_s_wait_asynccnt)
  __builtin_amdgcn_s_wait_asynccnt(0);
#else
  asm volatile("s_wait_asynccnt 0x0" ::: "memory");
#endif
}

// ---------------- K0: init stats ----------------
__global__ void k_init_stats(uint32_t* stats) {
  if (threadIdx.x == 0) {
    stats[0] = 0x7F7FFFFFu;  // minx bits (pos >= 0 so bit-min == float-min)
    stats[1] = 0x7F7FFFFFu;  // miny bits
    stats[2] = 0u;           // max |cart| bits (nonneg)
  }
}

// ---------------- K1: global min of pos (bitwise min, nonneg floats) --------
__global__ void k_pos_min(const float2* __restrict__ pos, int n, uint32_t* stats) {
  __shared__ uint32_t sx[256], sy[256];
  int t = threadIdx.x;
  uint32_t mx = 0x7F7FFFFFu, my = 0x7F7FFFFFu;
  for (int i = blockIdx.x * blockDim.x + t; i < n; i += gridDim.x * blockDim.x) {
    float2 p = pos[i];
    mx = min(mx, __float_as_uint(p.x));
    my = min(my, __float_as_uint(p.y));
  }
  sx[t] = mx; sy[t] = my;
  __syncthreads();
  for (int o = 128; o > 0; o >>= 1) {
    if (t < o) { sx[t] = min(sx[t], sx[t + o]); sy[t] = min(sy[t], sy[t + o]); }
    __syncthreads();
  }
  if (t == 0) { atomicMin(&stats[0], sx[0]); atomicMin(&stats[1], sy[0]); }
}

// ---------------- K2a: voxel hash per node + count + Q32.32 pos sums -------
__global__ void k_assign(const float2* __restrict__ pos, const int* __restrict__ batch,
                         int n, const uint32_t* __restrict__ stats,
                         uint32_t* __restrict__ cluster,
                         uint32_t* __restrict__ cnt,
                         unsigned long long* __restrict__ psum) {
  int i = blockIdx.x * blockDim.x + threadIdx.x;
  if (i >= n) return;
  float minx = __uint_as_float(stats[0]);
  float miny = __uint_as_float(stats[1]);
  float2 p = pos[i];
  int vx = (int)floorf((p.x - minx) * (float)GRIDV);   // (pos - start) / VOXEL_SIZE
  int vy = (int)floorf((p.y - miny) * (float)GRIDV);
  vx = min(max(vx, 0), GRIDV - 1);
  vy = min(max(vy, 0), GRIDV - 1);
  uint32_t c = (uint32_t)batch[i] * (GRIDV * GRIDV) + (uint32_t)vy * GRIDV + (uint32_t)vx;
  cluster[i] = c;
  atomicAdd(&cnt[c], 1u);
  // deterministic Q32.32 fixed-point accumulation (pos in [0,1))
  atomicAdd(&psum[2 * c + 0], (unsigned long long)(p.x * 4294967296.0));
  atomicAdd(&psum[2 * c + 1], (unsigned long long)(p.y * 4294967296.0));
}

// ---------------- generic 65536-bin exclusive scan (1 block, 1024 thr) -----
__global__ void k_scan(const uint32_t* __restrict__ hist, uint32_t* __restrict__ offsets) {
  __shared__ uint32_t part[1024];
  const int t = threadIdx.x;
  uint32_t s = 0;
  for (int i = 0; i < 64; ++i) s += hist[t * 64 + i];
  part[t] = s;
  __syncthreads();
  for (int o = 1; o < 1024; o <<= 1) {
    uint32_t add = (t >= o) ? part[t - o] : 0u;
    __syncthreads();
    part[t] += add;
    __syncthreads();
  }
  uint32_t run = (t > 0) ? part[t - 1] : 0u;
  for (int i = 0; i < 64; ++i) {
    offsets[t * 64 + i] = run;
    run += hist[t * 64 + i];
  }
  if (t == 1023) offsets[SSEG] = run;
}

// ---------------- K2c: group node indices by cluster ----------------
// (within-cluster order atomic-dependent; consumed only by order-invariant max)
__global__ void k_node_scatter(int n, const uint32_t* __restrict__ cluster,
                               const uint32_t* __restrict__ noffs,
                               uint32_t* __restrict__ ncur,
                               uint32_t* __restrict__ nodelist) {
  int i = blockIdx.x * blockDim.x + threadIdx.x;
  if (i >= n) return;
  uint32_t c = cluster[i];
  uint32_t p = noffs[c] + atomicAdd(&ncur[c], 1u);
  nodelist[p] = (uint32_t)i;
}

// ---------------- K2d: segmented feature max, one block per cluster --------
// Async-gathers member rows (256B each) into LDS via gfx1250 ASYNCcnt path,
// then register max-reduce -> writes x_pool directly. Zero global atomics.
__global__ void k_cluster_max(const float* __restrict__ x,
                              const uint32_t* __restrict__ nodelist,
                              const uint32_t* __restrict__ noffs,
                              float* __restrict__ xout) {
  __shared__ float    rows[LDSROWS * FDIM];   // 32 KB staged feature rows
  __shared__ uint32_t nl[LDSROWS];
  __shared__ float    red[256];
  const int s = blockIdx.x;
  const int t = threadIdx.x;
  const uint32_t b0 = noffs[s], b1 = noffs[s + 1];
  const int L = (int)(b1 - b0);
  const int f = t & 63, grp = t >> 6;         // 4 row-groups x 64 features
  float m = -3.402823466e+38f;

  const uint32_t lds_base = (uint32_t)(size_t)(&rows[0]);  // low 32 bits == LDS offset
  for (int base = 0; base < L; base += LDSROWS) {
    const int nb = min(LDSROWS, L - base);
    if (t < nb) nl[t] = nodelist[b0 + base + t];
    __syncthreads();
    const int tot = nb * (FDIM * 4 / 16);     // 16B chunks to move
    for (int i = t; i < tot; i += 256) {
      int r = i >> 4, j = i & 15;
      uint64_t gaddr = (uint64_t)(uintptr_t)x + ((size_t)nl[r] * (FDIM * 4) + (size_t)j * 16);
      async_load_b128(lds_base + (uint32_t)(r * (FDIM * 4) + j * 16), gaddr);
    }
    wait_async0();
    __syncthreads();
    for (int r = grp; r < nb; r += 4) m = fmaxf(m, rows[r * FDIM + f]);
    __syncthreads();                          // before next batch overwrites LDS
  }
  red[t] = m;
  __syncthreads();
  if (t < 64) {
    float v = fmaxf(fmaxf(red[t], red[t + 64]), fmaxf(red[t + 128], red[t + 192]));
    xout[(size_t)s * FDIM + t] = (L > 0) ? v : 0.0f;
  }
}

// ---------------- K3: edge -> 32-bit key (row<<16 | col) + row histogram ----
__global__ void k_edge_key(const int* __restrict__ ei, int E,
                           const uint32_t* __restrict__ cluster,
                           uint32_t* __restrict__ keys,
                           uint32_t* __restrict__ hist) {
  int e = blockIdx.x * blockDim.x + threadIdx.x;
  if (e >= E) return;
  __builtin_prefetch(&ei[min(e + 4096, E - 1)], 0, 0);      // global_prefetch_b8
  __builtin_prefetch(&ei[min(E + e + 4096, 2 * E - 1)], 0, 0);
  uint32_t r = cluster[ei[e]];
  uint32_t c = cluster[ei[E + e]];
  keys[e] = (r << 16) | c;
  atomicAdd(&hist[r], 1u);
}

// ---------------- K5: bucket-scatter keys by row ----------------
// (K6 fully sorts each row's key multiset -> final array deterministic)
__global__ void k_scatter(const uint32_t* __restrict__ keys, int E,
                          const uint32_t* __restrict__ offsets,
                          uint32_t* __restrict__ cursor,
                          uint32_t* __restrict__ skeys) {
  int e = blockIdx.x * blockDim.x + threadIdx.x;
  if (e >= E) return;
  uint32_t k = keys[e];
  uint32_t r = k >> 16;
  uint32_t p = offsets[r] + atomicAdd(&cursor[r], 1u);
  skeys[p] = k;
}

// ---------------- K6: per-row LDS odd-even sort (avg row len ~61) ----------
__global__ void k_rowsort(uint32_t* __restrict__ skeys,
                          const uint32_t* __restrict__ offsets) {
  __shared__ uint32_t buf[SORTCAP];
  const int row = blockIdx.x;
  const uint32_t s0 = offsets[row], s1 = offsets[row + 1];
  const int L = (int)(s1 - s0);
  if (L <= 1) return;
  const int n = min(L, SORTCAP);
  for (int i = threadIdx.x; i < n; i += blockDim.x) buf[i] = skeys[s0 + i];
  __syncthreads();
  for (int phase = 0; phase < n; ++phase) {
    for (int i = threadIdx.x;; i += blockDim.x) {
      int a = 2 * i + (phase & 1);
      if (a + 1 >= n) break;
      uint32_t u = buf[a], v = buf[a + 1];
      if (u > v) { buf[a] = v; buf[a + 1] = u; }
    }
    __syncthreads();
  }
  for (int i = threadIdx.x; i < n; i += blockDim.x) skeys[s0 + i] = buf[i];
}

// ---------------- K7: finalize pos_pool (x_pool already written) -----------
__global__ void k_finalize_pos(const uint32_t* __restrict__ cnt,
                               const unsigned long long* __restrict__ psum,
                               float* __restrict__ posout) {
  int i = blockIdx.x * blockDim.x + threadIdx.x;
  if (i >= SSEG * 2) return;
  int s = i >> 1;
  double denom = (double)max(cnt[s], 1u);
  posout[i] = (float)((double)psum[i] * (1.0 / 4294967296.0) / denom);
}

// ---------------- K8a: global max|cart| over valid coarse edges ------------
__global__ void k_edge_maxabs(const uint32_t* __restrict__ skeys, int E,
                              const float* __restrict__ posout,
                              uint32_t* __restrict__ stats) {
  int e = blockIdx.x * blockDim.x + threadIdx.x;
  if (e >= E) return;
  uint32_t k = skeys[e];
  bool first = (e == 0) || (k != skeys[e - 1]);
  uint32_t r = k >> 16, c = k & 0xFFFFu;
  if (first && r != c) {
    float cx = posout[2 * r + 0] - posout[2 * c + 0];
    float cy = posout[2 * r + 1] - posout[2 * c + 1];
    uint32_t bx = __float_as_uint(fabsf(cx));   // nonneg -> bits monotone
    uint32_t by = __float_as_uint(fabsf(cy));
    atomicMax(&stats[2], max(bx, by));
  }
}

// ---------------- K8b: write coarse edges + attrs + valid mask -------------
__global__ void k_edge_write(const uint32_t* __restrict__ skeys, int E,
                             const float* __restrict__ posout,
                             const uint32_t* __restrict__ stats,
                             float* __restrict__ erow, float* __restrict__ ecol,
                             float* __restrict__ attr, float* __restrict__ validf) {
  int e = blockIdx.x * blockDim.x + threadIdx.x;
  if (e >= E) return;
  uint32_t k = skeys[e];
  bool first = (e == 0) || (k != skeys[e - 1]);
  uint32_t r = k >> 16, c = k & 0xFFFFu;
  bool valid = first && (r != c);
  float maxv = __uint_as_float(stats[2]);
  float inv = 0.5f / fmaxf(maxv, 1e-30f);  // 1/(2*max)
  float ax = 0.0f, ay = 0.0f;
  if (valid) {
    float cx = posout[2 * r + 0] - posout[2 * c + 0];
    float cy = posout[2 * r + 1] - posout[2 * c + 1];
    ax = cx * inv + 0.5f;
    ay = cy * inv + 0.5f;
  }
  erow[e] = valid ? (float)r : 0.0f;
  ecol[e] = valid ? (float)c : 0.0f;
  attr[2 * e + 0] = ax;
  attr[2 * e + 1] = ay;
  validf[e] = valid ? 1.0f : 0.0f;
}

// ---------------- host launcher ----------------
extern "C" void kernel_launch(void* const* d_in, const int* in_sizes, int n_in,
                              void* d_out, int out_size, void* d_ws, size_t ws_size,
                              hipStream_t stream) {
  const float2* pos   = (const float2*)d_in[1];
  const float*  x     = (const float*)d_in[0];
  const int*    batch = (const int*)d_in[2];
  const int*    ei    = (const int*)d_in[3];
  const int N = in_sizes[2];
  const int E = in_sizes[3] / 2;

  // workspace carve-up (256B aligned)
  char* ws = (char*)d_ws;
  size_t off = 0;
  auto carve = [&](size_t bytes) { size_t o = off; off = (off + bytes + 255) & ~(size_t)255; return o; };
  size_t o_stats = carve(256);
  size_t o_clu   = carve((size_t)N * 4);
  size_t o_cnt   = carve((size_t)SSEG * 4);
  size_t o_noffs = carve((size_t)(SSEG + 1) * 4);
  size_t o_ncur  = carve((size_t)SSEG * 4);
  size_t o_nlist = carve((size_t)N * 4);
  size_t o_psum  = carve((size_t)SSEG * 2 * 8);
  size_t o_hist  = carve((size_t)SSEG * 4);
  size_t o_eoffs = carve((size_t)(SSEG + 1) * 4);
  size_t o_ecur  = carve((size_t)SSEG * 4);
  size_t o_keys  = carve((size_t)E * 4);
  size_t o_skey  = carve((size_t)E * 4);

  uint32_t*           stats = (uint32_t*)(ws + o_stats);
  uint32_t*           clu   = (uint32_t*)(ws + o_clu);
  uint32_t*           cnt   = (uint32_t*)(ws + o_cnt);
  uint32_t*           noffs = (uint32_t*)(ws + o_noffs);
  uint32_t*           ncur  = (uint32_t*)(ws + o_ncur);
  uint32_t*           nlist = (uint32_t*)(ws + o_nlist);
  unsigned long long* psum  = (unsigned long long*)(ws + o_psum);
  uint32_t*           hist  = (uint32_t*)(ws + o_hist);
  uint32_t*           eoffs = (uint32_t*)(ws + o_eoffs);
  uint32_t*           ecur  = (uint32_t*)(ws + o_ecur);
  uint32_t*           keys  = (uint32_t*)(ws + o_keys);
  uint32_t*           skeys = (uint32_t*)(ws + o_skey);

  // output carve-up (flat float, reference tuple return order)
  float* out    = (float*)d_out;
  float* xout   = out;                                  // [S, F]
  float* posout = xout + (size_t)SSEG * FDIM;           // [S, 2]
  float* erow   = posout + (size_t)SSEG * 2;            // [E]
  float* ecol   = erow + (size_t)E;                     // [E]
  float* attr   = ecol + (size_t)E;                     // [E, 2]
  float* validf = attr + (size_t)2 * E;                 // [E]

  hipMemsetAsync(d_ws, 0, off, stream);                 // zero accumulators (capture-safe)
  k_init_stats<<<1, 32, 0, stream>>>(stats);

  k_pos_min<<<1024, 256, 0, stream>>>(pos, N, stats);

  int nb = (N + 255) / 256;
  k_assign<<<nb, 256, 0, stream>>>(pos, batch, N, stats, clu, cnt, psum);
  k_scan<<<1, 1024, 0, stream>>>(cnt, noffs);
  k_node_scatter<<<nb, 256, 0, stream>>>(N, clu, noffs, ncur, nlist);
  k_cluster_max<<<SSEG, 256, 0, stream>>>(x, nlist, noffs, xout);

  int eb = (E + 255) / 256;
  k_edge_key<<<eb, 256, 0, stream>>>(ei, E, clu, keys, hist);
  k_scan<<<1, 1024, 0, stream>>>(hist, eoffs);
  k_scatter<<<eb, 256, 0, stream>>>(keys, E, eoffs, ecur, skeys);
  k_rowsort<<<SSEG, 256, 0, stream>>>(skeys, eoffs);

  k_finalize_pos<<<(SSEG * 2 + 255) / 256, 256, 0, stream>>>(cnt, psum, posout);
  k_edge_maxabs<<<eb, 256, 0, stream>>>(skeys, E, posout, stats);
  k_edge_write<<<eb, 256, 0, stream>>>(skeys, E, posout, stats, erow, ecol, attr, validf);
}